// GenerationGraph_47519518163579
// MI455X (gfx1250) — compile-verified
//
#include <hip/hip_runtime.h>
#include <hip/hip_bf16.h>
#include <math.h>

// Problem constants (from reference): P=32 preds, G=256 gallery, N=6890 SMPL verts,
// each cloud is the joint 2-person mesh: M = 2*N = 13780 points of 3 f32.
#define P_CNT 32
#define G_CNT 256
#define MV    13780           // vertices per cloud (divisible by 4 -> no WMMA K tail)
#define PT    4               // preds per workgroup in the error pass

typedef float v2f __attribute__((ext_vector_type(2)));
typedef float v8f __attribute__((ext_vector_type(8)));

// ---------------- workspace layout (floats) ----------------
// Kraw : 96 x 768 raw Gram matrix  (pred-rows x gt-cols)
#define KRAW_OFF 0
#define MUP_OFF  73728            // 32*3
#define VARP_OFF 73824            // 32
#define MUG_OFF  73856            // 256*3
#define RS_OFF   74624            // 8192 * 12  (R[9], scale, pad, pad)
#define ERR_OFF  172928           // 8192 pair errors
// total ~181k floats = ~0.7 MB of scratch

// ============================================================
// Kernel 1: per-cloud mean (+ optional centered variance)
// one workgroup (256 threads) per cloud; LDS tree reduction
// ============================================================
__global__ void cloud_stats_kernel(const float* __restrict__ pts,
                                   float* __restrict__ mu_out,
                                   float* __restrict__ var_out) {
  __shared__ float red[256 * 4];
  const int c = blockIdx.x;
  const float* base = pts + (size_t)c * MV * 3;
  float sx = 0.f, sy = 0.f, sz = 0.f, ssq = 0.f;
  for (int i = threadIdx.x; i < MV; i += 256) {
    float x = base[i * 3 + 0];
    float y = base[i * 3 + 1];
    float z = base[i * 3 + 2];
    sx += x; sy += y; sz += z;
    ssq += x * x + y * y + z * z;
  }
  red[threadIdx.x]        = sx;
  red[256 + threadIdx.x]  = sy;
  red[512 + threadIdx.x]  = sz;
  red[768 + threadIdx.x]  = ssq;
  __syncthreads();
  for (int s = 128; s > 0; s >>= 1) {
    if (threadIdx.x < s) {
      red[threadIdx.x]       += red[threadIdx.x + s];
      red[256 + threadIdx.x] += red[256 + threadIdx.x + s];
      red[512 + threadIdx.x] += red[512 + threadIdx.x + s];
      red[768 + threadIdx.x] += red[768 + threadIdx.x + s];
    }
    __syncthreads();
  }
  if (threadIdx.x == 0) {
    const float inv = 1.0f / (float)MV;
    float mx = red[0] * inv, my = red[256] * inv, mz = red[512] * inv;
    mu_out[c * 3 + 0] = mx;
    mu_out[c * 3 + 1] = my;
    mu_out[c * 3 + 2] = mz;
    if (var_out)  // var_p = sum||x||^2 - M*||mu||^2
      var_out[c] = red[768] - (float)MV * (mx * mx + my * my + mz * mz);
  }
}

// ============================================================
// Kernel 2: raw Gram GEMM via V_WMMA_F32_16X16X4_F32
//   A (96 x 13780):  A[p*3+c][i] = pred[p][i][c]
//   B (13780 x 768): B[i][g*3+c] = gt[g][i][c]
//   Kraw = A @ B ; one wave per 16x16 output tile, K-loop step 4
// A layout: lanes0-15 M=0..15 {V0=K0,V1=K1}; lanes16-31 {V0=K2,V1=K3}
// B layout: lanes0-15 N=0..15 {V0=K0,V1=K1}; lanes16-31 {V0=K2,V1=K3}
// D layout: VGPR j -> row j (lanes0-15) / row j+8 (lanes16-31), col=lane&15
// ============================================================
__global__ void pair_gram_wmma_kernel(const float* __restrict__ A,  // pred (P,MV,3)
                                      const float* __restrict__ B,  // gt   (G,MV,3)
                                      float* __restrict__ Kraw) {
  const int tileN = blockIdx.x;      // 0..47
  const int tileM = blockIdx.y;      // 0..5
  const int lane  = threadIdx.x;     // wave32
  const int half  = lane >> 4;
  const int l16   = lane & 15;

  const int arow = tileM * 16 + l16;  // 0..95
  const int bcol = tileN * 16 + l16;  // 0..767
  const float* aptr = A + (size_t)(arow / 3) * MV * 3 + (arow % 3);
  const float* bptr = B + (size_t)(bcol / 3) * MV * 3 + (bcol % 3);
  const int kofs = half * 2;

  v8f acc = {};
  for (int k0 = 0; k0 < MV; k0 += 4) {
    const int kb = k0 + kofs;
    v2f a, b;
    a.x = aptr[(size_t)kb * 3];
    a.y = aptr[(size_t)(kb + 1) * 3];
    b.x = bptr[(size_t)kb * 3];
    b.y = bptr[(size_t)(kb + 1) * 3];
    // D = A*B + C ; fp32 WMMA keeps full Procrustes precision
    acc = __builtin_amdgcn_wmma_f32_16x16x4_f32(
        /*neg_a=*/false, a, /*neg_b=*/false, b,
        /*c_mod=*/(short)0, acc, /*reuse_a=*/false, /*reuse_b=*/false);
  }
#pragma unroll
  for (int j = 0; j < 8; ++j) {
    const int row = tileM * 16 + half * 8 + j;
    Kraw[row * 768 + (tileN * 16 + l16)] = acc[j];
  }
}

// ============================================================
// Kernel 3: per-pair 3x3 Procrustes solve (one thread per pair)
//   K = Kraw - M*mu_p*mu_g^T ; Jacobi eigensolve of K^T K -> V, s^2
//   U = K V S^-1 ; R = V diag(1,1,d) U^T ; scale = (s0+s1+d*s2)/var_p
// ============================================================
__device__ __attribute__((always_inline)) inline
void jrot(float A[3][3], float V[3][3], int p, int q) {
  float apq = A[p][q];
  if (fabsf(apq) > 1e-30f) {
    float tau = (A[q][q] - A[p][p]) / (2.f * apq);
    float t = (tau >= 0.f ? 1.f : -1.f) / (fabsf(tau) + sqrtf(1.f + tau * tau));
    float c = 1.f / sqrtf(1.f + t * t);
    float s = t * c;
#pragma unroll
    for (int k = 0; k < 3; ++k) {
      float akp = A[k][p], akq = A[k][q];
      A[k][p] = c * akp - s * akq;
      A[k][q] = s * akp + c * akq;
    }
#pragma unroll
    for (int k = 0; k < 3; ++k) {
      float apk = A[p][k], aqk = A[q][k];
      A[p][k] = c * apk - s * aqk;
      A[q][k] = s * apk + c * aqk;
    }
#pragma unroll
    for (int k = 0; k < 3; ++k) {
      float vkp = V[k][p], vkq = V[k][q];
      V[k][p] = c * vkp - s * vkq;
      V[k][q] = s * vkp + c * vkq;
    }
  }
}

__device__ __attribute__((always_inline)) inline float det3(const float m[3][3]) {
  return m[0][0] * (m[1][1] * m[2][2] - m[1][2] * m[2][1])
       - m[0][1] * (m[1][0] * m[2][2] - m[1][2] * m[2][0])
       + m[0][2] * (m[1][0] * m[2][1] - m[1][1] * m[2][0]);
}

__global__ void procrustes_svd_kernel(const float* __restrict__ Kraw,
                                      const float* __restrict__ muP,
                                      const float* __restrict__ varP,
                                      const float* __restrict__ muG,
                                      float* __restrict__ Rs) {
  const int t = blockIdx.x * blockDim.x + threadIdx.x;
  if (t >= P_CNT * G_CNT) return;
  const int p = t >> 8;       // /256
  const int g = t & 255;

  float K[3][3];
#pragma unroll
  for (int a = 0; a < 3; ++a)
#pragma unroll
    for (int b = 0; b < 3; ++b)
      K[a][b] = Kraw[(p * 3 + a) * 768 + (g * 3 + b)]
              - (float)MV * muP[p * 3 + a] * muG[g * 3 + b];

  float A[3][3], V[3][3];
#pragma unroll
  for (int i = 0; i < 3; ++i)
#pragma unroll
    for (int j = 0; j < 3; ++j) {
      A[i][j] = K[0][i] * K[0][j] + K[1][i] * K[1][j] + K[2][i] * K[2][j];
      V[i][j] = (i == j) ? 1.f : 0.f;
    }
#pragma unroll
  for (int sweep = 0; sweep < 6; ++sweep) {
    jrot(A, V, 0, 1);
    jrot(A, V, 0, 2);
    jrot(A, V, 1, 2);
  }
  float d[3] = {A[0][0], A[1][1], A[2][2]};
  auto swapcols = [&](int i, int j) {
    float td = d[i]; d[i] = d[j]; d[j] = td;
#pragma unroll
    for (int k = 0; k < 3; ++k) { float tv = V[k][i]; V[k][i] = V[k][j]; V[k][j] = tv; }
  };
  if (d[0] < d[1]) swapcols(0, 1);
  if (d[0] < d[2]) swapcols(0, 2);
  if (d[1] < d[2]) swapcols(1, 2);

  float sv[3];
#pragma unroll
  for (int k = 0; k < 3; ++k) sv[k] = sqrtf(fmaxf(d[k], 0.f));

  float U[3][3];
#pragma unroll
  for (int k = 0; k < 3; ++k) {
    float inv = 1.f / fmaxf(sv[k], 1e-20f);
#pragma unroll
    for (int a = 0; a < 3; ++a)
      U[a][k] = (K[a][0] * V[0][k] + K[a][1] * V[1][k] + K[a][2] * V[2][k]) * inv;
  }
  const float dsgn = (det3(U) * det3(V) < 0.f) ? -1.f : 1.f;
  const float D[3] = {1.f, 1.f, dsgn};

  float* out = Rs + (size_t)t * 12;
#pragma unroll
  for (int i = 0; i < 3; ++i)
#pragma unroll
    for (int j = 0; j < 3; ++j)
      out[i * 3 + j] = V[i][0] * D[0] * U[j][0] + V[i][1] * D[1] * U[j][1]
                     + V[i][2] * D[2] * U[j][2];
  out[9]  = (sv[0] + sv[1] + dsgn * sv[2]) / fmaxf(varP[p], 1e-20f);
  out[10] = 0.f;
  out[11] = 0.f;
}

// ============================================================
// Kernel 4: per-pair mean vertex error (dominant pass; L2-resident)
// one WG of 256 threads covers PT preds x 1 gt  -> gt slice read once per PT pairs
// err_i = || s*R*(x-mu_p) - (z-mu_g) ||
// ============================================================
__global__ void pair_err_kernel(const float* __restrict__ pred,
                                const float* __restrict__ gt,
                                const float* __restrict__ muP,
                                const float* __restrict__ muG,
                                const float* __restrict__ Rs,
                                float* __restrict__ errOut) {
  __shared__ float prm[PT * 13];   // per pred: R[9], s, mu_p[3]
  __shared__ float mg[3];
  __shared__ float red[256];
  const int g   = blockIdx.x & 255;
  const int pb  = blockIdx.x >> 8;     // 0..(P_CNT/PT-1)
  const int tid = threadIdx.x;

  if (tid < PT) {
    const int p = pb * PT + tid;
    const float* r = Rs + (size_t)(p * G_CNT + g) * 12;
#pragma unroll
    for (int k = 0; k < 10; ++k) prm[tid * 13 + k] = r[k];
    prm[tid * 13 + 10] = muP[p * 3 + 0];
    prm[tid * 13 + 11] = muP[p * 3 + 1];
    prm[tid * 13 + 12] = muP[p * 3 + 2];
  }
  if (tid == PT) { mg[0] = muG[g * 3]; mg[1] = muG[g * 3 + 1]; mg[2] = muG[g * 3 + 2]; }
  __syncthreads();

  // hoist parameters to registers (constant across the vertex loop)
  float q[PT][13];
#pragma unroll
  for (int j = 0; j < PT; ++j)
#pragma unroll
    for (int k = 0; k < 13; ++k) q[j][k] = prm[j * 13 + k];
  const float mgx = mg[0], mgy = mg[1], mgz = mg[2];

  float acc[PT];
#pragma unroll
  for (int j = 0; j < PT; ++j) acc[j] = 0.f;

  const float* gbase = gt + (size_t)g * MV * 3;
  for (int i = tid; i < MV; i += 256) {
    __builtin_prefetch(gbase + (size_t)(i + 256) * 3, 0, 1);  // global_prefetch_b8
    const float zx = gbase[i * 3 + 0] - mgx;
    const float zy = gbase[i * 3 + 1] - mgy;
    const float zz = gbase[i * 3 + 2] - mgz;
#pragma unroll
    for (int j = 0; j < PT; ++j) {
      const float* xb = pred + (size_t)(pb * PT + j) * MV * 3 + (size_t)i * 3;
      const float xx = xb[0] - q[j][10];
      const float xy = xb[1] - q[j][11];
      const float xz = xb[2] - q[j][12];
      const float s  = q[j][9];
      const float yx = s * (q[j][0] * xx + q[j][1] * xy + q[j][2] * xz) - zx;
      const float yy = s * (q[j][3] * xx + q[j][4] * xy + q[j][5] * xz) - zy;
      const float yz = s * (q[j][6] * xx + q[j][7] * xy + q[j][8] * xz) - zz;
      acc[j] += sqrtf(yx * yx + yy * yy + yz * yz);
    }
  }

  for (int j = 0; j < PT; ++j) {
    red[tid] = acc[j];
    __syncthreads();
    for (int s = 128; s > 0; s >>= 1) {
      if (tid < s) red[tid] += red[tid + s];
      __syncthreads();
    }
    if (tid == 0)
      errOut[(pb * PT + j) * G_CNT + g] = red[0] * (1.0f / (float)MV);
    __syncthreads();
  }
}

// ============================================================
// Kernel 5: argmin over gallery per prediction
// ============================================================
__global__ void argmin_kernel(const float* __restrict__ err,
                              int* __restrict__ mapIdx,
                              float* __restrict__ minErr) {
  const int p = blockIdx.x * blockDim.x + threadIdx.x;
  if (p >= P_CNT) return;
  const float* e = err + p * G_CNT;
  float best = e[0];
  int bi = 0;
  for (int gI = 1; gI < G_CNT; ++gI) {
    float v = e[gI];
    if (v < best) { best = v; bi = gI; }  // first-min, matches jnp.argmin
  }
  mapIdx[p] = bi;
  minErr[p] = best;
}

// ============================================================
extern "C" void kernel_launch(void* const* d_in, const int* in_sizes, int n_in,
                              void* d_out, int out_size, void* d_ws, size_t ws_size,
                              hipStream_t stream) {
  (void)in_sizes; (void)n_in; (void)out_size; (void)ws_size;
  const float* pred = (const float*)d_in[0];   // (32, 2, 6890, 3) f32
  const float* gtp  = (const float*)d_in[1];   // (256, 2, 6890, 3) f32
  float* ws   = (float*)d_ws;
  float* Kraw = ws + KRAW_OFF;
  float* muP  = ws + MUP_OFF;
  float* varP = ws + VARP_OFF;
  float* muG  = ws + MUG_OFF;
  float* Rs   = ws + RS_OFF;
  float* perr = ws + ERR_OFF;

  cloud_stats_kernel<<<P_CNT, 256, 0, stream>>>(pred, muP, varP);
  cloud_stats_kernel<<<G_CNT, 256, 0, stream>>>(gtp, muG, nullptr);
  pair_gram_wmma_kernel<<<dim3(48, 6), 32, 0, stream>>>(pred, gtp, Kraw);
  procrustes_svd_kernel<<<(P_CNT * G_CNT) / 256, 256, 0, stream>>>(Kraw, muP, varP, muG, Rs);
  pair_err_kernel<<<(P_CNT / PT) * G_CNT, 256, 0, stream>>>(pred, gtp, muP, muG, Rs, perr);
  argmin_kernel<<<1, 32, 0, stream>>>(perr, (int*)d_out, (float*)d_out + P_CNT);
}